// cheb_poly_gcn_50354196578314
// MI455X (gfx1250) — compile-verified
//
#include <hip/hip_runtime.h>

typedef float v2f __attribute__((ext_vector_type(2)));
typedef float v8f __attribute__((ext_vector_type(8)));

#define B_   16
#define N_   512
#define CIN  64
#define COUT 64
#define T_   64
#define KCH  3
#define TM   (T_ * N_)     // 32768 = flattened (t,m)
#define COLS (COUT * T_)   // 4096  = flattened (o,t)
#define NN   (N_ * N_)

// ---------------------------------------------------------------------------
// CDNA5 async global->LDS copy (GLOBAL_LOAD_ASYNC_TO_LDS_B128, ASYNCcnt).
// Builtin signature (per hipcc diagnostic): (v4i AS1*, v4i AS3*, imm, imm)
// where v4i = int __attribute__((vector_size(16))).
// Guarded: falls back to sync copy if the builtins aren't exposed.
// ---------------------------------------------------------------------------
#if defined(__AMDGCN__) &&                                        \
    __has_builtin(__builtin_amdgcn_global_load_async_to_lds_b128) && \
    __has_builtin(__builtin_amdgcn_s_wait_asynccnt)
#define HAVE_ASYNC_LDS 1
#else
#define HAVE_ASYNC_LDS 0
#endif

#if HAVE_ASYNC_LDS
typedef int v4i_t __attribute__((vector_size(16)));
typedef __attribute__((address_space(1))) v4i_t* as1_v4i;
typedef __attribute__((address_space(3))) v4i_t* as3_v4i;
#endif

__device__ __forceinline__ void async_copy16(const float* gsrc, float* ldst) {
#if HAVE_ASYNC_LDS
  __builtin_amdgcn_global_load_async_to_lds_b128((as1_v4i)gsrc, (as3_v4i)ldst,
                                                 0, 0);
#else
  *(float4*)ldst = *(const float4*)gsrc;
#endif
}

__device__ __forceinline__ void wait_async0() {
#if HAVE_ASYNC_LDS
  __builtin_amdgcn_s_wait_asynccnt(0);
#endif
}

// D = A(16x4) * B(4x16) + C, full f32 via CDNA5 WMMA
__device__ __forceinline__ v8f wmma4(v2f a, v2f b, v8f c) {
  return __builtin_amdgcn_wmma_f32_16x16x4_f32(false, a, false, b, (short)0, c,
                                               false, false);
}

// ---------------------------------------------------------------------------
// Kernel 1: T2[b] = 2 * La[b] @ La[b] - I      (per-batch 512x512x512 GEMM)
// Tile: 64 (n) x 128 (q), K-step 32. 8 waves, each 32x32.
// B tile stored TRANSPOSED in LDS so each 16x4 fragment is an adjacent f32
// pair -> single ds_load_b64, no packing movs.
// ---------------------------------------------------------------------------
__global__ __launch_bounds__(256) void cheb2_kernel(const float* __restrict__ La,
                                                    float* __restrict__ T2) {
  __shared__ float As[64][36];    // A tile [n][m]
  __shared__ float Bs[128][36];   // B tile transposed: [q][m]
  const int b = blockIdx.z;
  const int nBase = blockIdx.y * 64;
  const int qBase = blockIdx.x * 128;
  const float* Lb = La + (size_t)b * NN;
  const int tid = threadIdx.x;
  const int wid = tid >> 5, lane = tid & 31, half = lane >> 4, l16 = lane & 15;
  const int waveM = wid & 1;   // over n (2 x 32)
  const int waveN = wid >> 1;  // over q (4 x 32)

  v8f acc[2][2];
  for (int i = 0; i < 2; ++i)
    for (int j = 0; j < 2; ++j)
      for (int e = 0; e < 8; ++e) acc[i][j][e] = 0.0f;

  for (int m0 = 0; m0 < N_; m0 += 32) {
    // A tile: 64 rows x 32 m, contiguous 16B chunks -> async to LDS
    for (int i = tid; i < 64 * 8; i += 256) {
      int row = i >> 3, m4 = (i & 7) << 2;
      async_copy16(Lb + (size_t)(nBase + row) * N_ + m0 + m4, &As[row][m4]);
    }
    // B tile: load [m][q] rows coalesced, scatter transposed to Bs[q][m]
    for (int i = tid; i < 32 * 32; i += 256) {
      int row = i >> 5, q4 = (i & 31) << 2;
      float4 v = *(const float4*)(Lb + (size_t)(m0 + row) * N_ + qBase + q4);
      Bs[q4 + 0][row] = v.x;
      Bs[q4 + 1][row] = v.y;
      Bs[q4 + 2][row] = v.z;
      Bs[q4 + 3][row] = v.w;
    }
    wait_async0();
    __syncthreads();
    for (int kk0 = 0; kk0 < 32; kk0 += 4) {
      v2f a[2], bb[2];
      for (int mt = 0; mt < 2; ++mt) {
        int row = waveM * 32 + mt * 16 + l16;
        a[mt].x = As[row][kk0 + 2 * half];
        a[mt].y = As[row][kk0 + 2 * half + 1];
      }
      for (int nt = 0; nt < 2; ++nt) {
        int q = waveN * 32 + nt * 16 + l16;
        bb[nt].x = Bs[q][kk0 + 2 * half];
        bb[nt].y = Bs[q][kk0 + 2 * half + 1];
      }
      for (int mt = 0; mt < 2; ++mt)
        for (int nt = 0; nt < 2; ++nt)
          acc[mt][nt] = wmma4(a[mt], bb[nt], acc[mt][nt]);
    }
    __syncthreads();
  }
  float* Tb = T2 + (size_t)b * NN;
  for (int mt = 0; mt < 2; ++mt)
    for (int nt = 0; nt < 2; ++nt)
      for (int v = 0; v < 8; ++v) {
        int n = nBase + waveM * 32 + mt * 16 + v + 8 * half;
        int q = qBase + waveN * 32 + nt * 16 + l16;
        float val = 2.0f * acc[mt][nt][v];
        if (n == q) val -= 1.0f;
        Tb[(size_t)n * N_ + q] = val;
      }
}

// ---------------------------------------------------------------------------
// Kernel 2 (per batch): G_k (64 x 32768) = W_k (64x64) @ X (64 x 32768)
// X = x[b] viewed as [c_in, T*N]. W_k[o,c] = W[o, 3c+k]. Grid: (tm/256, k).
// ---------------------------------------------------------------------------
__global__ __launch_bounds__(256) void gmat_kernel(const float* __restrict__ xb,
                                                   const float* __restrict__ W,
                                                   float* __restrict__ G) {
  __shared__ float Ws[64][65];
  const int k = blockIdx.y;
  const int tmBase = blockIdx.x * 256;
  const int tid = threadIdx.x;
  for (int i = tid; i < 64 * 64; i += 256) {
    int o = i >> 6, c = i & 63;
    Ws[o][c] = W[o * (KCH * CIN) + c * KCH + k];
  }
  __syncthreads();
  const int wid = tid >> 5, lane = tid & 31, half = lane >> 4, l16 = lane & 15;
  const int tmW = tmBase + wid * 32;   // each wave owns 32 tm columns

  v8f acc[4][2];
  for (int i = 0; i < 4; ++i)
    for (int j = 0; j < 2; ++j)
      for (int e = 0; e < 8; ++e) acc[i][j][e] = 0.0f;

  for (int c0 = 0; c0 < CIN; c0 += 4) {
    v2f a[4];
    for (int ot = 0; ot < 4; ++ot) {
      int row = ot * 16 + l16;
      a[ot].x = Ws[row][c0 + 2 * half];
      a[ot].y = Ws[row][c0 + 2 * half + 1];
    }
    v2f bx[2];
    for (int st = 0; st < 2; ++st) {
      int tm = tmW + st * 16 + l16;
      bx[st].x = xb[(size_t)(c0 + 2 * half) * TM + tm];
      bx[st].y = xb[(size_t)(c0 + 2 * half + 1) * TM + tm];
    }
    for (int ot = 0; ot < 4; ++ot)
      for (int st = 0; st < 2; ++st)
        acc[ot][st] = wmma4(a[ot], bx[st], acc[ot][st]);
  }
  for (int ot = 0; ot < 4; ++ot)
    for (int st = 0; st < 2; ++st)
      for (int v = 0; v < 8; ++v) {
        int o = ot * 16 + v + 8 * half;
        int tm = tmW + st * 16 + l16;
        G[((size_t)k * COUT + o) * TM + tm] = acc[ot][st][v];
      }
}

// ---------------------------------------------------------------------------
// Kernel 3 (per batch): out[b] (4096 x 512) = G0 + G1 @ La^T + G2 @ T2^T
// G_k flat is [col=(o*64+t)][m]; out flat is [col][n]. Tile: 64 col x 128 n.
// Double-buffered LDS; next K-slab streamed global->LDS via async copies
// while current slab feeds the WMMAs.
// ---------------------------------------------------------------------------
__global__ __launch_bounds__(256) void gconv_out_kernel(const float* __restrict__ G,
                                                        const float* __restrict__ Lb,
                                                        const float* __restrict__ Tb,
                                                        float* __restrict__ outb) {
  __shared__ float Gs[2][2][64][36];    // [buf][s] A tiles: G1,G2  [col][m]
  __shared__ float Lt[2][2][128][36];   // [buf][s] B tiles: La,T2  [n][m]
  const int nBase = blockIdx.x * 128;
  const int colBase = blockIdx.y * 64;
  const int tid = threadIdx.x;
  const int wid = tid >> 5, lane = tid & 31, half = lane >> 4, l16 = lane & 15;
  const int waveM = wid & 1;   // over col (2 x 32)
  const int waveN = wid >> 1;  // over n   (4 x 32)

  auto issue_tiles = [&](int buf, int m0) {
    for (int i = tid; i < 2 * 64 * 8; i += 256) {   // G1/G2 tiles
      int s = i >> 9, rem = i & 511;
      int row = rem >> 3, m4 = (rem & 7) << 2;
      async_copy16(G + ((size_t)(s + 1) * COUT) * TM +
                       (size_t)(colBase + row) * N_ + m0 + m4,
                   &Gs[buf][s][row][m4]);
    }
    for (int i = tid; i < 2 * 128 * 8; i += 256) {  // La/T2 tiles
      int s = i >> 10, rem = i & 1023;
      int row = rem >> 3, m4 = (rem & 7) << 2;
      const float* src = (s == 0) ? Lb : Tb;
      async_copy16(src + (size_t)(nBase + row) * N_ + m0 + m4,
                   &Lt[buf][s][row][m4]);
    }
  };

  v8f acc[2][2];
  for (int i = 0; i < 2; ++i)
    for (int j = 0; j < 2; ++j)
      for (int e = 0; e < 8; ++e) acc[i][j][e] = 0.0f;

  issue_tiles(0, 0);
  int buf = 0;
  for (int m0 = 0; m0 < N_; m0 += 32) {
    wait_async0();        // current buf's copies have landed in LDS
    __syncthreads();      // every wave done with the other buf -> reusable
    if (m0 + 32 < N_) issue_tiles(buf ^ 1, m0 + 32);   // prefetch next slab

    for (int kk0 = 0; kk0 < 32; kk0 += 4) {
      v2f aG[2][2], bL[2][2];
      for (int s = 0; s < 2; ++s)
        for (int ct = 0; ct < 2; ++ct) {
          int row = waveM * 32 + ct * 16 + l16;
          aG[s][ct].x = Gs[buf][s][row][kk0 + 2 * half];
          aG[s][ct].y = Gs[buf][s][row][kk0 + 2 * half + 1];
        }
      for (int s = 0; s < 2; ++s)
        for (int nt = 0; nt < 2; ++nt) {
          int row = waveN * 32 + nt * 16 + l16;
          bL[s][nt].x = Lt[buf][s][row][kk0 + 2 * half];
          bL[s][nt].y = Lt[buf][s][row][kk0 + 2 * half + 1];
        }
      for (int ct = 0; ct < 2; ++ct)
        for (int nt = 0; nt < 2; ++nt) {
          acc[ct][nt] = wmma4(aG[0][ct], bL[0][nt], acc[ct][nt]);
          acc[ct][nt] = wmma4(aG[1][ct], bL[1][nt], acc[ct][nt]);
        }
    }
    buf ^= 1;
  }
  // epilogue: add identity term (G0[col][n]) and store out[b][col][n]
  for (int ct = 0; ct < 2; ++ct)
    for (int nt = 0; nt < 2; ++nt)
      for (int v = 0; v < 8; ++v) {
        int col = colBase + waveM * 32 + ct * 16 + v + 8 * half;
        int n = nBase + waveN * 32 + nt * 16 + l16;
        outb[(size_t)col * N_ + n] = acc[ct][nt][v] + G[(size_t)col * N_ + n];
      }
}

// ---------------------------------------------------------------------------
extern "C" void kernel_launch(void* const* d_in, const int* in_sizes, int n_in,
                              void* d_out, int out_size, void* d_ws, size_t ws_size,
                              hipStream_t stream) {
  (void)in_sizes; (void)n_in; (void)out_size; (void)ws_size;
  const float* x  = (const float*)d_in[0];  // [16,64,64,512]
  const float* La = (const float*)d_in[1];  // [16,512,512]
  const float* W  = (const float*)d_in[2];  // [64,192]
  float* out = (float*)d_out;               // [16,64,64,512]
  float* T2 = (float*)d_ws;                 // 16*512*512 floats (16 MB)
  float* G  = T2 + (size_t)B_ * NN;         // 3*64*32768 floats (24 MB), reused per b

  // Chebyshev T2 for all batches
  cheb2_kernel<<<dim3(N_ / 128, N_ / 64, B_), 256, 0, stream>>>(La, T2);

  // Per batch: channel-mix GEMM, then graph-conv GEMMs (stream order serializes)
  for (int b = 0; b < B_; ++b) {
    gmat_kernel<<<dim3(TM / 256, KCH), 256, 0, stream>>>(
        x + (size_t)b * CIN * TM, W, G);
    gconv_out_kernel<<<dim3(N_ / 128, COLS / 64), 256, 0, stream>>>(
        G, La + (size_t)b * NN, T2 + (size_t)b * NN, out + (size_t)b * COLS * N_);
  }
}